// HamiltonianClassifier_19361712571096
// MI455X (gfx1250) — compile-verified
//
#include <hip/hip_runtime.h>
#include <hip/hip_bf16.h>
#include <math.h>

typedef float v2f __attribute__((ext_vector_type(2)));
typedef float v8f __attribute__((ext_vector_type(8)));

#define BATCH 32
#define SEQ   512
#define DMODEL 768
#define QDIM  1024
#define EPS2  1e-24f

// ---------------- kernel 0: zero the ev accumulator ----------------
__global__ void init_ev_kernel(float* __restrict__ evacc) {
    if (threadIdx.x < BATCH) evacc[threadIdx.x] = 0.0f;
}

// ---------------- kernel 1: sent = normalize(mean_s x) ----------------
// one block per batch, 256 threads, each thread owns d, d+256, d+512
__global__ __launch_bounds__(256) void sent_kernel(const float* __restrict__ x,
                                                   float* __restrict__ sentn) {
    int b = blockIdx.x;
    int t = threadIdx.x;
    const float* xb = x + (size_t)b * SEQ * DMODEL;
    float a0 = 0.f, a1 = 0.f, a2 = 0.f;
    for (int s = 0; s < SEQ; ++s) {
        const float* row = xb + (size_t)s * DMODEL;
        a0 += row[t];
        a1 += row[t + 256];
        a2 += row[t + 512];
    }
    const float inv = 1.0f / (float)SEQ;
    a0 *= inv; a1 *= inv; a2 *= inv;

    __shared__ float red[256];
    red[t] = a0 * a0 + a1 * a1 + a2 * a2;
    __syncthreads();
    for (int off = 128; off > 0; off >>= 1) {
        if (t < off) red[t] += red[t + off];
        __syncthreads();
    }
    float invd = 1.0f / sqrtf(red[0] + EPS2);
    float* dst = sentn + (size_t)b * QDIM;
    dst[t]       = a0 * invd;
    dst[t + 256] = a1 * invd;
    dst[t + 512] = a2 * invd;
    dst[t + 768] = 0.0f;   // pad region
}

// ---------------- kernel 2: state = sentn @ U ----------------
// one block per batch; sentn (only 768 nonzero entries) staged in LDS
__global__ __launch_bounds__(256) void state_kernel(const float* __restrict__ sentn,
                                                    const float* __restrict__ U,
                                                    float* __restrict__ state) {
    int b = blockIdx.x;
    int t = threadIdx.x;
    __shared__ float sv[DMODEL];
    for (int i = t; i < DMODEL; i += 256) sv[i] = sentn[(size_t)b * QDIM + i];
    __syncthreads();
    float acc0 = 0.f, acc1 = 0.f, acc2 = 0.f, acc3 = 0.f;
    for (int i = 0; i < DMODEL; ++i) {   // rows >= 768 of sentn are zero
        float s = sv[i];
        const float* ur = U + (size_t)i * QDIM;
        acc0 = fmaf(s, ur[t],       acc0);
        acc1 = fmaf(s, ur[t + 256], acc1);
        acc2 = fmaf(s, ur[t + 512], acc2);
        acc3 = fmaf(s, ur[t + 768], acc3);
    }
    float* dst = state + (size_t)b * QDIM;
    dst[t]       = acc0;
    dst[t + 256] = acc1;
    dst[t + 512] = acc2;
    dst[t + 768] = acc3;
}

// ---------------- kernel 3: ham[b] = (pos .* xb)^T @ xb / len[b]  (f32 WMMA) ----
// LDS tiles are TRANSPOSED ([column][k], even padded stride) so each WMMA
// fragment is one naturally-aligned ds_load_b64 into an even VGPR pair:
// no repacking moves in the hot loop.
#define TILE 128
#define KC   32
#define LDK  34    // padded k-stride (even -> 8B-aligned b64; 34 mod 64 walk is
                   // conflict-free across 16 lanes)

__global__ __launch_bounds__(256) void ham_kernel(const float* __restrict__ x,
                                                  const int*   __restrict__ lens,
                                                  const float* __restrict__ bias,
                                                  const float* __restrict__ pos,
                                                  float*       __restrict__ ham) {
    __shared__ float At[TILE * LDK];   // pos-weighted xb, i-columns (17408 B)
    __shared__ float Bt[TILE * LDK];   // xb, j-columns            (17408 B)

    const int b  = blockIdx.z;
    const int i0 = blockIdx.y * TILE;
    const int j0 = blockIdx.x * TILE;
    const int len = lens[b];
    const float invlen = 1.0f / (float)len;

    const int tid  = threadIdx.x;
    const int lane = tid & 31;
    const int wave = tid >> 5;        // 0..7
    const int wr   = wave >> 2;       // 0..1 -> 64-row half
    const int wc   = wave & 3;        // 0..3 -> 32-col quarter

    v8f acc[4][2];
    for (int a = 0; a < 4; ++a)
        for (int c = 0; c < 2; ++c)
            for (int q = 0; q < 8; ++q) acc[a][c][q] = 0.0f;

    const float* xb = x + (size_t)b * SEQ * DMODEL;
    const int cg = lane;          // 32 column-groups of float4 = 128 floats/row
    const int r0 = tid >> 5;      // 8 row groups
    const int m16 = lane & 15;
    const int klane = (lane >> 4) << 1;   // lanes 16-31 hold K+2 / K+3

    for (int k0 = 0; k0 < SEQ; k0 += KC) {
        // stage KC x 128 tiles into LDS, transposed, bias+mask+pos applied
        for (int rr = 0; rr < KC; rr += 8) {
            int kl = r0 + rr;             // local k
            int k  = k0 + kl;
            float m = (k < len) ? 1.0f : 0.0f;
            float p = pos[k];
            const float* src = xb + (size_t)k * DMODEL;
            int c4 = cg * 4;

            float4 va = *(const float4*)(src + i0 + c4);
            float4 ba = *(const float4*)(bias + i0 + c4);
            At[(c4 + 0) * LDK + kl] = p * (va.x + m * ba.x);
            At[(c4 + 1) * LDK + kl] = p * (va.y + m * ba.y);
            At[(c4 + 2) * LDK + kl] = p * (va.z + m * ba.z);
            At[(c4 + 3) * LDK + kl] = p * (va.w + m * ba.w);

            float4 vb = *(const float4*)(src + j0 + c4);
            float4 bb = *(const float4*)(bias + j0 + c4);
            Bt[(c4 + 0) * LDK + kl] = vb.x + m * bb.x;
            Bt[(c4 + 1) * LDK + kl] = vb.y + m * bb.y;
            Bt[(c4 + 2) * LDK + kl] = vb.z + m * bb.z;
            Bt[(c4 + 3) * LDK + kl] = vb.w + m * bb.w;

            // prefetch next K-chunk row (lowers to global_prefetch_b8)
            if (k0 + KC < SEQ)
                __builtin_prefetch(src + (size_t)KC * DMODEL + i0 + c4, 0, 1);
        }
        __syncthreads();

        // WMMA sweep over this K chunk: pure ds_load_b64 + v_wmma
        #pragma unroll
        for (int kk = 0; kk < KC; kk += 4) {
            int kh = kk + klane;
            v2f afr[4], bfr[2];
            #pragma unroll
            for (int a = 0; a < 4; ++a) {
                int irow = wr * 64 + a * 16 + m16;
                afr[a] = *(const v2f*)(At + irow * LDK + kh);
            }
            #pragma unroll
            for (int c = 0; c < 2; ++c) {
                int jcol = wc * 32 + c * 16 + m16;
                bfr[c] = *(const v2f*)(Bt + jcol * LDK + kh);
            }
            #pragma unroll
            for (int a = 0; a < 4; ++a)
                #pragma unroll
                for (int c = 0; c < 2; ++c)
                    acc[a][c] = __builtin_amdgcn_wmma_f32_16x16x4_f32(
                        false, afr[a], false, bfr[c], (short)0, acc[a][c],
                        false, false);
        }
        __syncthreads();
    }

    // store C: VGPR v holds row M=v (lanes 0-15) / M=v+8 (lanes 16-31)
    const int hi = lane >> 4;
    for (int a = 0; a < 4; ++a) {
        for (int c = 0; c < 2; ++c) {
            int jg = j0 + wc * 32 + c * 16 + m16;
            for (int v = 0; v < 8; ++v) {
                int ig = i0 + wr * 64 + a * 16 + v + hi * 8;
                ham[((size_t)b * DMODEL + ig) * DMODEL + jg] = acc[a][c][v] * invlen;
            }
        }
    }
}

// ---------------- kernel 4: fused cross-batch normalize + ev contraction ------
__global__ __launch_bounds__(256) void ev_kernel(const float* __restrict__ ham,
                                                 const float* __restrict__ state,
                                                 float* __restrict__ evacc) {
    __shared__ float evloc[BATCH];
    int t = threadIdx.x;
    if (t < BATCH) evloc[t] = 0.0f;
    __syncthreads();

    size_t idx = (size_t)blockIdx.x * blockDim.x + t;   // over 768*768 pairs
    if (idx < (size_t)DMODEL * DMODEL) {
        int i = (int)(idx / DMODEL);
        int j = (int)(idx % DMODEL);
        float v[BATCH];
        float ss = EPS2;
        #pragma unroll
        for (int b = 0; b < BATCH; ++b) {
            float h = ham[((size_t)b * DMODEL + i) * DMODEL + j];
            v[b] = h;
            ss = fmaf(h, h, ss);
        }
        float invd = 1.0f / sqrtf(ss);
        #pragma unroll
        for (int b = 0; b < BATCH; ++b) {
            float c = state[(size_t)b * QDIM + i] * v[b] *
                      state[(size_t)b * QDIM + j] * invd;
            atomicAdd(&evloc[b], c);
        }
    }
    __syncthreads();
    if (t < BATCH) atomicAdd(&evacc[t], evloc[t]);
}

// ---------------- kernel 5: sigmoid ----------------
__global__ void sigmoid_kernel(const float* __restrict__ evacc,
                               float* __restrict__ out) {
    int t = threadIdx.x;
    if (t < BATCH) out[t] = 1.0f / (1.0f + expf(-evacc[t]));
}

extern "C" void kernel_launch(void* const* d_in, const int* in_sizes, int n_in,
                              void* d_out, int out_size, void* d_ws, size_t ws_size,
                              hipStream_t stream) {
    const float* x    = (const float*)d_in[0];   // [32,512,768]
    const int*   lens = (const int*)  d_in[1];   // [32]
    const float* bias = (const float*)d_in[2];   // [768]
    const float* pos  = (const float*)d_in[3];   // [1024]
    const float* U    = (const float*)d_in[4];   // [1024,1024]

    float* out = (float*)d_out;          // ev: out[0..31], state: out[32..32+32767]
    float* state = out + BATCH;

    // workspace layout
    char* ws = (char*)d_ws;
    float* ham   = (float*)ws;                                        // 75.5 MB
    float* sentn = (float*)(ws + (size_t)BATCH * DMODEL * DMODEL * 4);// 128 KB
    float* evacc = sentn + (size_t)BATCH * QDIM;                      // 128 B

    init_ev_kernel<<<1, 32, 0, stream>>>(evacc);
    sent_kernel<<<BATCH, 256, 0, stream>>>(x, sentn);
    state_kernel<<<BATCH, 256, 0, stream>>>(sentn, U, state);

    dim3 hgrid(DMODEL / TILE, DMODEL / TILE, BATCH);   // (6,6,32)
    ham_kernel<<<hgrid, 256, 0, stream>>>(x, lens, bias, pos, ham);

    int npairs = DMODEL * DMODEL;
    ev_kernel<<<(npairs + 255) / 256, 256, 0, stream>>>(ham, state, evacc);
    sigmoid_kernel<<<1, 32, 0, stream>>>(evacc, out);
}